// GAT_78176994721828
// MI455X (gfx1250) — compile-verified
//
#include <hip/hip_runtime.h>

// GATv2 2-layer forward for MI455X (gfx1250, wave32).
// GEMMs via V_WMMA_F32_16X16X4_F32 (exact f32 — GEMM is not the bottleneck;
// the edge gather/scatter phase is, and it lives in the 192MB L2).

#define N_NODES 50000
#define N_EDGES 800000
#define EP (N_EDGES + N_NODES)      // 850000 edges incl self-loops
#define IN_DIM 128
#define C1 256                      // per-side channels, layer 1 (4 heads * 64)
#define D1 512                      // combined [xl|xr] layer-1 row width
#define C2 40
#define D2 80                       // combined [xl|xr] layer-2 row width
#define NEG_SLOPE 0.2f
#define ROWT ((N_NODES + 15) / 16)  // 3125 row tiles (exact)

typedef float v2f __attribute__((ext_vector_type(2)));
typedef float v8f __attribute__((ext_vector_type(8)));

// order-preserving float <-> uint map for atomicMax-based segment max
__device__ __forceinline__ unsigned fmap(float f) {
  unsigned b = __float_as_uint(f);
  return (b & 0x80000000u) ? ~b : (b | 0x80000000u);
}
__device__ __forceinline__ float funmap(unsigned u) {
  return (u & 0x80000000u) ? __uint_as_float(u ^ 0x80000000u)
                           : __uint_as_float(~u);
}

__global__ void k_zero(float* __restrict__ p, size_t n) {
  size_t i = (size_t)blockIdx.x * blockDim.x + threadIdx.x;
  size_t stride = (size_t)gridDim.x * blockDim.x;
  for (; i < n; i += stride) p[i] = 0.0f;
}

// ---------------- GEMM 1: xw1[N,512] = x[N,128] @ [Wl1|Wr1] ----------------
__global__ void k_gemm1(const float* __restrict__ x, const float* __restrict__ Wl,
                        const float* __restrict__ Wr, float* __restrict__ xw) {
  __shared__ float lw[IN_DIM * 16];           // 8 KB weight tile (K x 16)
  const int n0 = blockIdx.x * 16;
  for (int i = threadIdx.x; i < IN_DIM * 16; i += blockDim.x) {
    int k = i >> 4, j = i & 15, n = n0 + j;
    lw[i] = (n < C1) ? Wl[(size_t)k * C1 + n] : Wr[(size_t)k * C1 + (n - C1)];
  }
  __syncthreads();
  const int wave = threadIdx.x >> 5, lane = threadIdx.x & 31;
  const int rt = blockIdx.y * 4 + wave;
  if (rt >= ROWT) return;                     // wave-uniform guard
  const int r0 = rt * 16;
  const int row = lane & 15, kh = lane >> 4;  // A: M=row, K-half per lane group
  const float* xr = x + (size_t)(r0 + row) * IN_DIM;
  v8f acc = {};
#pragma unroll
  for (int k = 0; k < IN_DIM; k += 4) {
    v2f a, b;
    a.x = xr[k + 2 * kh];
    a.y = xr[k + 2 * kh + 1];
    b.x = lw[(k + 2 * kh) * 16 + row];
    b.y = lw[(k + 2 * kh + 1) * 16 + row];
    acc = __builtin_amdgcn_wmma_f32_16x16x4_f32(false, a, false, b,
                                                (short)0, acc, false, false);
  }
  float* op = xw + (size_t)(r0 + kh * 8) * D1 + n0 + row;
#pragma unroll
  for (int i = 0; i < 8; ++i) op[(size_t)i * D1] = acc[i];
}

// ---------------- GEMM 2: xw2[N,80] = h[N,256] @ [Wl2|Wr2] ----------------
__global__ void k_gemm2(const float* __restrict__ h, const float* __restrict__ Wl,
                        const float* __restrict__ Wr, float* __restrict__ xw) {
  __shared__ float lw[C1 * 16];               // 16 KB weight tile (K x 16)
  const int n0 = blockIdx.x * 16;
  for (int i = threadIdx.x; i < C1 * 16; i += blockDim.x) {
    int k = i >> 4, j = i & 15, n = n0 + j;
    lw[i] = (n < C2) ? Wl[(size_t)k * C2 + n] : Wr[(size_t)k * C2 + (n - C2)];
  }
  __syncthreads();
  const int wave = threadIdx.x >> 5, lane = threadIdx.x & 31;
  const int rt = blockIdx.y * 4 + wave;
  if (rt >= ROWT) return;
  const int r0 = rt * 16;
  const int row = lane & 15, kh = lane >> 4;
  const float* hr = h + (size_t)(r0 + row) * C1;
  v8f acc = {};
#pragma unroll
  for (int k = 0; k < C1; k += 4) {
    v2f a, b;
    a.x = hr[k + 2 * kh];
    a.y = hr[k + 2 * kh + 1];
    b.x = lw[(k + 2 * kh) * 16 + row];
    b.y = lw[(k + 2 * kh + 1) * 16 + row];
    acc = __builtin_amdgcn_wmma_f32_16x16x4_f32(false, a, false, b,
                                                (short)0, acc, false, false);
  }
  float* op = xw + (size_t)(r0 + kh * 8) * D2 + n0 + row;
#pragma unroll
  for (int i = 0; i < 8; ++i) op[(size_t)i * D2] = acc[i];
}

__device__ __forceinline__ void edge_sd(const int* __restrict__ ei, int edge,
                                        int& s, int& d) {
  if (edge < N_EDGES) { s = ei[edge]; d = ei[N_EDGES + edge]; }
  else { s = edge - N_EDGES; d = s; }
}

// ---------- layer 1 edge logits + segment max (one wave32 per edge) --------
__global__ void k_edge_att1(const int* __restrict__ ei, const float* __restrict__ xw,
                            const float* __restrict__ att, float* __restrict__ e1,
                            unsigned* __restrict__ m1) {
  const int edge = blockIdx.x * 8 + (threadIdx.x >> 5);
  if (edge >= EP) return;
  const int lane = threadIdx.x & 31;
  int s, d; edge_sd(ei, edge, s, d);
  const float* xl = xw + (size_t)s * D1;        // source transform (cols 0..255)
  const float* xr = xw + (size_t)d * D1 + C1;   // target transform (cols 256..511)
  float hs[4] = {0.f, 0.f, 0.f, 0.f};
#pragma unroll
  for (int k = 0; k < 8; ++k) {                 // c = lane + 32k; head = k>>1 (const)
    int c = lane + 32 * k;
    float v = xl[c] + xr[c];
    v = (v > 0.f) ? v : NEG_SLOPE * v;
    hs[k >> 1] += v * att[c];
  }
#pragma unroll
  for (int h = 0; h < 4; ++h)
#pragma unroll
    for (int off = 16; off > 0; off >>= 1) hs[h] += __shfl_xor(hs[h], off, 32);
  if (lane == 0) {
#pragma unroll
    for (int h = 0; h < 4; ++h) {
      e1[(size_t)edge * 4 + h] = hs[h];
      atomicMax(&m1[(size_t)d * 4 + h], fmap(hs[h]));
    }
  }
}

// ---------- layer 1: a = exp(e - max), segment sum ----------
__global__ void k_edge_exp1(const int* __restrict__ ei, float* __restrict__ e1,
                            const unsigned* __restrict__ m1, float* __restrict__ den1) {
  size_t t = (size_t)blockIdx.x * blockDim.x + threadIdx.x;
  if (t >= (size_t)EP * 4) return;
  int edge = (int)(t >> 2), h = (int)(t & 3);
  int s, d; edge_sd(ei, edge, s, d); (void)s;
  float a = expf(e1[t] - funmap(m1[(size_t)d * 4 + h]));
  e1[t] = a;
  atomicAdd(&den1[(size_t)d * 4 + h], a);
}

// ---------- layer 1 aggregation: agg[dst] += alpha * xl[src] ----------
__global__ void k_edge_agg1(const int* __restrict__ ei, const float* __restrict__ xw,
                            const float* __restrict__ e1, const float* __restrict__ den1,
                            float* __restrict__ agg) {
  const int edge = blockIdx.x * 8 + (threadIdx.x >> 5);
  if (edge >= EP) return;
  const int lane = threadIdx.x & 31;
  int s, d; edge_sd(ei, edge, s, d);
  float alpha[4];
#pragma unroll
  for (int h = 0; h < 4; ++h)
    alpha[h] = e1[(size_t)edge * 4 + h] / (den1[(size_t)d * 4 + h] + 1e-16f);
  const float* xl = xw + (size_t)s * D1;
  float* ag = agg + (size_t)d * C1;
#pragma unroll
  for (int k = 0; k < 8; ++k) {
    int c = lane + 32 * k;
    atomicAdd(&ag[c], alpha[k >> 1] * xl[c]);
  }
}

// ---------- batchnorm stats (column-parallel, coalesced) ----------
__global__ void k_bn_stats(const float* __restrict__ agg, const float* __restrict__ b1,
                           float* __restrict__ bnsum, float* __restrict__ bnss) {
  const int c = threadIdx.x;
  const int r0 = blockIdx.x * 250;
  float s = 0.f, ss = 0.f;
  for (int r = r0; r < r0 + 250; ++r) {
    float v = agg[(size_t)r * C1 + c] + b1[c];
    s += v; ss += v * v;
  }
  atomicAdd(&bnsum[c], s);
  atomicAdd(&bnss[c], ss);
}

// ---------- batchnorm normalize + ELU, in place ----------
__global__ void k_bn_elu(float* __restrict__ agg, const float* __restrict__ b1,
                         const float* __restrict__ g1, const float* __restrict__ be1,
                         const float* __restrict__ bnsum, const float* __restrict__ bnss) {
  size_t t = (size_t)blockIdx.x * blockDim.x + threadIdx.x;
  if (t >= (size_t)N_NODES * C1) return;
  int c = (int)(t & (C1 - 1));
  const float invn = 1.0f / (float)N_NODES;
  float mu = bnsum[c] * invn;
  float var = bnss[c] * invn - mu * mu;
  float v = agg[t] + b1[c];
  v = (v - mu) * rsqrtf(var + 1e-5f) * g1[c] + be1[c];
  agg[t] = (v > 0.f) ? v : expm1f(v);
}

// ---------- layer 2 edge logits + segment max ----------
__global__ void k_edge_att2(const int* __restrict__ ei, const float* __restrict__ xw,
                            const float* __restrict__ att, float* __restrict__ e2,
                            unsigned* __restrict__ m2) {
  const int edge = blockIdx.x * 8 + (threadIdx.x >> 5);
  if (edge >= EP) return;
  const int lane = threadIdx.x & 31;
  int s, d; edge_sd(ei, edge, s, d);
  const float* xl = xw + (size_t)s * D2;
  const float* xr = xw + (size_t)d * D2 + C2;
  float acc = 0.f;
  {
    int c = lane;                                // c in [0,32) < 40 always
    float v = xl[c] + xr[c];
    v = (v > 0.f) ? v : NEG_SLOPE * v;
    acc += v * att[c];
  }
  if (lane < C2 - 32) {                          // c in [32,40)
    int c = lane + 32;
    float v = xl[c] + xr[c];
    v = (v > 0.f) ? v : NEG_SLOPE * v;
    acc += v * att[c];
  }
#pragma unroll
  for (int off = 16; off > 0; off >>= 1) acc += __shfl_xor(acc, off, 32);
  if (lane == 0) {
    e2[edge] = acc;
    atomicMax(&m2[d], fmap(acc));
  }
}

__global__ void k_edge_exp2(const int* __restrict__ ei, float* __restrict__ e2,
                            const unsigned* __restrict__ m2, float* __restrict__ den2) {
  int edge = blockIdx.x * blockDim.x + threadIdx.x;
  if (edge >= EP) return;
  int s, d; edge_sd(ei, edge, s, d); (void)s;
  float a = expf(e2[edge] - funmap(m2[d]));
  e2[edge] = a;
  atomicAdd(&den2[d], a);
}

__global__ void k_edge_agg2(const int* __restrict__ ei, const float* __restrict__ xw,
                            const float* __restrict__ e2, const float* __restrict__ den2,
                            float* __restrict__ out) {
  const int edge = blockIdx.x * 8 + (threadIdx.x >> 5);
  if (edge >= EP) return;
  const int lane = threadIdx.x & 31;
  int s, d; edge_sd(ei, edge, s, d);
  float alpha = e2[edge] / (den2[d] + 1e-16f);
  const float* xl = xw + (size_t)s * D2;
  float* op = out + (size_t)d * C2;
  atomicAdd(&op[lane], alpha * xl[lane]);
  if (lane < C2 - 32) atomicAdd(&op[lane + 32], alpha * xl[lane + 32]);
}

__global__ void k_final(float* __restrict__ out, const float* __restrict__ b2) {
  size_t t = (size_t)blockIdx.x * blockDim.x + threadIdx.x;
  if (t >= (size_t)N_NODES * C2) return;
  out[t] += b2[t % C2];
}

extern "C" void kernel_launch(void* const* d_in, const int* in_sizes, int n_in,
                              void* d_out, int out_size, void* d_ws, size_t ws_size,
                              hipStream_t stream) {
  const float* x    = (const float*)d_in[0];
  const int*   ei   = (const int*)d_in[1];
  const float* Wl1  = (const float*)d_in[2];
  const float* Wr1  = (const float*)d_in[3];
  const float* att1 = (const float*)d_in[4];
  const float* b1   = (const float*)d_in[5];
  const float* g1   = (const float*)d_in[6];
  const float* be1  = (const float*)d_in[7];
  const float* Wl2  = (const float*)d_in[8];
  const float* Wr2  = (const float*)d_in[9];
  const float* att2 = (const float*)d_in[10];
  const float* b2   = (const float*)d_in[11];
  float* out = (float*)d_out;

  // workspace layout (floats)
  float*    ws    = (float*)d_ws;
  float*    xw1   = ws;                                   // N*512
  float*    e1    = xw1 + (size_t)N_NODES * D1;           // EP*4
  unsigned* m1    = (unsigned*)(e1 + (size_t)EP * 4);     // N*4
  float*    den1  = (float*)m1 + (size_t)N_NODES * 4;     // N*4
  float*    agg1  = den1 + (size_t)N_NODES * 4;           // N*256
  float*    bnsum = agg1 + (size_t)N_NODES * C1;          // 256
  float*    bnss  = bnsum + C1;                           // 256
  float*    xw2   = bnss + C1;                            // N*80
  float*    e2    = xw2 + (size_t)N_NODES * D2;           // EP
  unsigned* m2    = (unsigned*)(e2 + (size_t)EP);         // N
  float*    den2  = (float*)m2 + N_NODES;                 // N

  const int ewaves = (EP + 7) / 8;
  const dim3 blk256(256), blk128(128);

  // zero: [m1 .. bnss] contiguous, [m2,den2] contiguous, d_out
  k_zero<<<2048, blk256, 0, stream>>>((float*)m1,
      (size_t)N_NODES * (4 + 4 + C1) + 2 * C1);
  k_zero<<<256, blk256, 0, stream>>>((float*)m2, (size_t)2 * N_NODES);
  k_zero<<<2048, blk256, 0, stream>>>(out, (size_t)N_NODES * C2);

  // layer 1
  k_gemm1<<<dim3(D1 / 16, (ROWT + 3) / 4), blk128, 0, stream>>>(x, Wl1, Wr1, xw1);
  k_edge_att1<<<ewaves, blk256, 0, stream>>>(ei, xw1, att1, e1, m1);
  k_edge_exp1<<<((size_t)EP * 4 + 255) / 256, blk256, 0, stream>>>(ei, e1, m1, den1);
  k_edge_agg1<<<ewaves, blk256, 0, stream>>>(ei, xw1, e1, den1, agg1);
  k_bn_stats<<<200, blk256, 0, stream>>>(agg1, b1, bnsum, bnss);
  k_bn_elu<<<(N_NODES * C1) / 256, blk256, 0, stream>>>(agg1, b1, g1, be1, bnsum, bnss);

  // layer 2
  k_gemm2<<<dim3(D2 / 16, (ROWT + 3) / 4), blk128, 0, stream>>>(agg1, Wl2, Wr2, xw2);
  k_edge_att2<<<ewaves, blk256, 0, stream>>>(ei, xw2, att2, e2, m2);
  k_edge_exp2<<<(EP + 255) / 256, blk256, 0, stream>>>(ei, e2, m2, den2);
  k_edge_agg2<<<ewaves, blk256, 0, stream>>>(ei, xw2, e2, den2, out);
  k_final<<<((size_t)N_NODES * C2 + 255) / 256, blk256, 0, stream>>>(out, b2);
}